// GCNEncoder_47545287966961
// MI455X (gfx1250) — compile-verified
//
#include <hip/hip_runtime.h>

typedef __attribute__((ext_vector_type(2))) float v2f;
typedef __attribute__((ext_vector_type(8))) float v8f;

#define CH 128   // feature width for both fused passes (layer2 = [mu|logstd])

// ---------------- degree / normalization ----------------
__global__ void deg_init_kernel(float* __restrict__ deg, int n) {
  int i = blockIdx.x * blockDim.x + threadIdx.x;
  if (i < n) deg[i] = 1.0f;  // self-loop contributes 1 to every node's degree
}

__global__ void deg_count_kernel(const int* __restrict__ col, float* __restrict__ deg, int e) {
  int i = blockIdx.x * blockDim.x + threadIdx.x;
  if (i < e) unsafeAtomicAdd(&deg[col[i]], 1.0f);
}

__global__ void deg_rsqrt_kernel(float* __restrict__ deg, int n) {
  int i = blockIdx.x * blockDim.x + threadIdx.x;
  if (i < n) deg[i] = rsqrtf(deg[i]);  // deg >= 1 always (self loop)
}

// ---------------- weight concat: Wcat[k][c] = c<64 ? Wmu[k][c] : Wls[k][c-64] ----------------
__global__ void concat_w_kernel(const float* __restrict__ Wmu, const float* __restrict__ Wls,
                                float* __restrict__ Wcat) {
  int i = blockIdx.x * blockDim.x + threadIdx.x;   // 0 .. 128*128
  if (i < 128 * 128) {
    int k = i >> 7, c = i & 127;
    Wcat[i] = (c < 64) ? Wmu[k * 64 + c] : Wls[k * 64 + (c - 64)];
  }
}

// ---------------- fp32 WMMA GEMM: C[M,128] = A[M,128] x B[128,128] ----------------
// One wave computes one 16x16 output tile via 32 chained V_WMMA_F32_16X16X4_F32.
// Requires M % 16 == 0 (N=100000 = 6250*16).
__global__ __launch_bounds__(256) void gemm_wmma_f32_kernel(const float* __restrict__ A,
                                                            const float* __restrict__ B,
                                                            float* __restrict__ C, int M) {
  int wid  = (blockIdx.x * blockDim.x + threadIdx.x) >> 5;
  int lane = threadIdx.x & 31;
  int tm = wid >> 3;        // 8 column tiles (128/16) per tile-row
  int tn = wid & 7;
  if (tm * 16 >= M) return;

  int hi = lane >> 4;       // 0: lanes 0-15, 1: lanes 16-31
  int lo = lane & 15;

  const float* arow = A + (tm * 16 + lo) * CH;   // A: lane holds row M=lo; K pair (2*hi, 2*hi+1)
  const float* bcol = B + tn * 16 + lo;          // B: lane holds col N=lo; K pair (2*hi, 2*hi+1)

  v8f acc = {};
#pragma unroll 4
  for (int k = 0; k < CH; k += 4) {
    v2f a, b;
    a.x = arow[k + 2 * hi];
    a.y = arow[k + 2 * hi + 1];
    b.x = bcol[(k + 2 * hi) * CH];
    b.y = bcol[(k + 2 * hi + 1) * CH];
    acc = __builtin_amdgcn_wmma_f32_16x16x4_f32(
        /*neg_a=*/false, a, /*neg_b=*/false, b,
        /*c_mod=*/(short)0, acc, /*reuse_a=*/false, /*reuse_b=*/false);
  }

  // C/D layout: VGPR v -> row (v + 8*hi), col = lo
  float* crow = C + (tm * 16 + 8 * hi) * CH + tn * 16 + lo;
#pragma unroll
  for (int v = 0; v < 8; ++v) crow[v * CH] = acc[v];
}

// ---------------- scatter: self-loop init agg[i] = dinv[i]^2 * h[i] ----------------
__global__ void scatter_init_kernel(const float* __restrict__ h, const float* __restrict__ dinv,
                                    float* __restrict__ agg, int total) {
  int i = blockIdx.x * blockDim.x + threadIdx.x;
  if (i < total) {
    float d = dinv[i >> 7];
    agg[i] = d * d * h[i];
  }
}

// ---------------- scatter: one wave per edge, 4 channels per lane ----------------
// Edge id is wave-uniform: force it into an SGPR so index/dinv loads scalarize
// (constant-cache path) and only the float4 gather + 4 atomics use the vector pipe.
__global__ __launch_bounds__(256) void scatter_edge_kernel(const int* __restrict__ row,
                                                           const int* __restrict__ col,
                                                           const float* __restrict__ dinv,
                                                           const float* __restrict__ h,
                                                           float* __restrict__ agg, int e) {
  int wid = (blockIdx.x * blockDim.x + threadIdx.x) >> 5;
  wid = __builtin_amdgcn_readfirstlane(wid);          // wave-uniform edge id
  if (wid >= e) return;
  int lane = threadIdx.x & 31;
  int r = __builtin_amdgcn_readfirstlane(row[wid]);   // uniform source node
  int c = __builtin_amdgcn_readfirstlane(col[wid]);   // uniform target node
  float norm = dinv[r] * dinv[c];                     // uniform edge weight
  const float4 hv = *reinterpret_cast<const float4*>(h + r * CH + lane * 4);
  float* dst = agg + c * CH + lane * 4;
  unsafeAtomicAdd(dst + 0, norm * hv.x);
  unsafeAtomicAdd(dst + 1, norm * hv.y);
  unsafeAtomicAdd(dst + 2, norm * hv.z);
  unsafeAtomicAdd(dst + 3, norm * hv.w);
}

// ---------------- h = relu(agg + b1), in place ----------------
__global__ void bias_relu_kernel(float* __restrict__ agg, const float* __restrict__ b, int total) {
  int i = blockIdx.x * blockDim.x + threadIdx.x;
  if (i < total) {
    float v = agg[i] + b[i & 127];
    agg[i] = v > 0.0f ? v : 0.0f;
  }
}

// ---------------- split fused agg2 into mu / logstd with biases ----------------
// Output is write-once: non-temporal stores keep L2 free for the feature buffers.
__global__ void final_split_kernel(const float* __restrict__ agg, const float* __restrict__ bmu,
                                   const float* __restrict__ bls, float* __restrict__ out,
                                   int n) {
  int i = blockIdx.x * blockDim.x + threadIdx.x;   // over n*128
  if (i < n * CH) {
    int node = i >> 7, ch = i & 127;
    if (ch < 64)
      __builtin_nontemporal_store(agg[i] + bmu[ch], &out[node * 64 + ch]);            // mu
    else
      __builtin_nontemporal_store(agg[i] + bls[ch - 64],
                                  &out[n * 64 + node * 64 + (ch - 64)]);              // logstd
  }
}

extern "C" void kernel_launch(void* const* d_in, const int* in_sizes, int n_in,
                              void* d_out, int out_size, void* d_ws, size_t ws_size,
                              hipStream_t stream) {
  // inputs: x, edge_index, batch, W1, b1, Wmu, bmu, Wls, bls
  const float* x    = (const float*)d_in[0];
  const int*   eidx = (const int*)d_in[1];
  const float* W1   = (const float*)d_in[3];
  const float* b1   = (const float*)d_in[4];
  const float* Wmu  = (const float*)d_in[5];
  const float* bmu  = (const float*)d_in[6];
  const float* Wls  = (const float*)d_in[7];
  const float* bls  = (const float*)d_in[8];
  float* out = (float*)d_out;

  const int N = in_sizes[0] / CH;       // 100000
  const int E = in_sizes[1] / 2;        // 1600000
  const int* erow = eidx;               // sources
  const int* ecol = eidx + E;           // targets

  // workspace layout (floats): dinv[N] | Wcat[128*128] | buf0[N*128] | buf1[N*128]
  float* dinv = (float*)d_ws;
  float* Wcat = dinv + N;
  float* buf0 = Wcat + 128 * 128;
  float* buf1 = buf0 + (size_t)N * CH;

  const int total = N * CH;
  const int TB = 256;
  const int gN     = (N + TB - 1) / TB;
  const int gE     = (E + TB - 1) / TB;
  const int gTot   = (total + TB - 1) / TB;
  const int gEdgeW = (E * 32 + TB - 1) / TB;            // one wave32 per edge
  const int mtiles = (N + 15) / 16;
  const int gGemm  = (mtiles * 8 * 32 + TB - 1) / TB;   // 8 col-tiles, 32 lanes per wave

  // normalization
  deg_init_kernel<<<gN, TB, 0, stream>>>(dinv, N);
  deg_count_kernel<<<gE, TB, 0, stream>>>(ecol, dinv, E);
  deg_rsqrt_kernel<<<gN, TB, 0, stream>>>(dinv, N);
  concat_w_kernel<<<(128 * 128 + TB - 1) / TB, TB, 0, stream>>>(Wmu, Wls, Wcat);

  // layer 1: h0 = x @ W1 ; agg1 = D^-1/2 A D^-1/2 h0 ; h = relu(agg1 + b1)
  gemm_wmma_f32_kernel<<<gGemm, TB, 0, stream>>>(x, W1, buf0, N);
  scatter_init_kernel<<<gTot, TB, 0, stream>>>(buf0, dinv, buf1, total);
  scatter_edge_kernel<<<gEdgeW, TB, 0, stream>>>(erow, ecol, dinv, buf0, buf1, E);
  bias_relu_kernel<<<gTot, TB, 0, stream>>>(buf1, b1, total);

  // layer 2 (fused mu|logstd): h2 = h @ [Wmu|Wls] ; agg2 = norm-scatter(h2)
  gemm_wmma_f32_kernel<<<gGemm, TB, 0, stream>>>(buf1, Wcat, buf0, N);
  scatter_init_kernel<<<gTot, TB, 0, stream>>>(buf0, dinv, buf1, total);
  scatter_edge_kernel<<<gEdgeW, TB, 0, stream>>>(erow, ecol, dinv, buf0, buf1, E);

  // split + biases into d_out: [mu (N*64) | logstd (N*64)]
  final_split_kernel<<<gTot, TB, 0, stream>>>(buf1, bmu, bls, out, N);
}